// HashGridEncoder_49718541418694
// MI455X (gfx1250) — compile-verified
//
#include <hip/hip_runtime.h>
#include <math.h>

namespace {

constexpr unsigned NPTS  = 524288u;          // N_POINTS
constexpr unsigned LVLS  = 16u;              // L
constexpr unsigned TBL   = 524288u;          // T (power of two)
constexpr unsigned TMASK = TBL - 1u;
constexpr unsigned FHL   = 5u;               // first hash level
constexpr unsigned P1    = 2654435761u;
constexpr unsigned P2    = 805459861u;

// Tiled-level resolutions and exact cumulative table offsets (entries).
// res: 16,23,31,43,59 (59^3 = 205379 <= T; 81^3 > T -> hash from level 5)
__constant__ unsigned c_res[5]  = {16u, 23u, 31u, 43u, 59u};
__constant__ unsigned c_off[16] = {
    0u,       4096u,    16263u,   46054u,   125561u,
    330940u,  855228u,  1379516u, 1903804u, 2428092u,
    2952380u, 3476668u, 4000956u, 4525244u, 5049532u, 5573820u};

typedef float v2f __attribute__((ext_vector_type(2)));

struct alignas(16) LevelMeta {
  float scale;      // float32(16 * 2^(7l/15) - 1)
  unsigned off;     // table offset (entries)
  unsigned res;     // resolution (tiled levels only)
  unsigned res2;    // res*res
};

__global__ __launch_bounds__(256) void hashgrid_fwd(
    const float* __restrict__ pos,      // (N, 3)
    const float* __restrict__ latents,  // (6098108, 2)
    const float* __restrict__ bound_p,  // scalar
    float* __restrict__ out)            // (N, 32) + 1 (tv)
{
  __shared__ LevelMeta s_meta[16];

  const unsigned t = threadIdx.x;

  // 16 threads/block compute per-level metadata once; everyone else gets it
  // with a single ds_load_b128. The f64 exp2 keeps the float32 scale cast
  // bit-compatible with the reference's float64 grid_meta() path
  // (scale14 = 1481.0036 sits 0.003 above an integer -- precision matters).
  if (t < 16u) {
    const float scale = (float)(16.0 * exp2((double)t * (7.0 / 15.0)) - 1.0);
    const unsigned r = (t < FHL) ? c_res[t] : 0u;
    s_meta[t].scale = scale;
    s_meta[t].off   = c_off[t];
    s_meta[t].res   = r;
    s_meta[t].res2  = r * r;
  }
  __syncthreads();

  const unsigned tid = blockIdx.x * 256u + t;
  const unsigned p   = tid >> 4;    // point index: 16 lanes per point
  const unsigned l   = t & 15u;     // level index

  const LevelMeta m = s_meta[l];

  const float bound = bound_p[0];

  // Streaming input: 16 lanes share a point's cacheline; NT keeps L2 for the table.
  const float px = __builtin_nontemporal_load(pos + 3u * p + 0u);
  const float py = __builtin_nontemporal_load(pos + 3u * p + 1u);
  const float pz = __builtin_nontemporal_load(pos + 3u * p + 2u);

  // posn = (pos + bound) / (2*bound). For bound == 1 (harness value) the
  // divide is by an exact power of two -> multiply by 0.5f is bit-identical
  // and saves 3 full-precision divide sequences. Uniform branch.
  const float nx = px + bound, ny = py + bound, nz = pz + bound;
  float posx, posy, posz;
  if (bound == 1.0f) {
    posx = nx * 0.5f; posy = ny * 0.5f; posz = nz * 0.5f;
  } else {
    const float inv2b = 2.0f * bound;
    posx = nx / inv2b; posy = ny / inv2b; posz = nz / inv2b;
  }

  const float sx = posx * m.scale + 0.5f;
  const float sy = posy * m.scale + 0.5f;
  const float sz = posz * m.scale + 0.5f;

  const float fx = floorf(sx), fy = floorf(sy), fz = floorf(sz);
  const float tx = sx - fx,    ty = sy - fy,    tz = sz - fz;
  const unsigned bx = (unsigned)fx, by = (unsigned)fy, bz = (unsigned)fz;

  // Corner c: dim0(x) = (c>>2)&1, dim1(y) = (c>>1)&1, dim2(z) = c&1
  unsigned idx[8];
  if (l < FHL) {
    const unsigned base = bx + by * m.res + bz * m.res2;
#pragma unroll
    for (int c = 0; c < 8; ++c) {
      const unsigned v = base + ((unsigned)(c >> 2) & 1u)
                              + (((unsigned)(c >> 1) & 1u) * m.res)
                              + (((unsigned)c & 1u) * m.res2);
      idx[c] = (v & TMASK) + m.off;
    }
  } else {
#pragma unroll
    for (int c = 0; c < 8; ++c) {
      const unsigned x = bx + ((unsigned)(c >> 2) & 1u);
      const unsigned y = by + ((unsigned)(c >> 1) & 1u);
      const unsigned z = bz + ((unsigned)c & 1u);
      idx[c] = ((x ^ (y * P1) ^ (z * P2)) & TMASK) + m.off;
    }
  }

  // 8 independent float2 gathers from the L2-resident table: issue all,
  // then one wait -> maximal memory-level parallelism per thread.
  const v2f* __restrict__ lt = (const v2f*)latents;
  v2f lat[8];
#pragma unroll
  for (int c = 0; c < 8; ++c) lat[c] = lt[idx[c]];

  const float wx[2] = {1.0f - tx, tx};
  const float wy[2] = {1.0f - ty, ty};
  const float wz[2] = {1.0f - tz, tz};

  v2f acc = {0.0f, 0.0f};
#pragma unroll
  for (int c = 0; c < 8; ++c) {
    const float w = wx[(c >> 2) & 1] * wy[(c >> 1) & 1] * wz[c & 1];
    acc += w * lat[c];
  }

  // (N, L*F) row-major: point p owns out[32p .. 32p+31]; lane layout makes
  // each wave's 32 float2 stores one contiguous 256B burst. NT: output is
  // write-once, keep it out of L2.
  __builtin_nontemporal_store(acc, ((v2f*)out) + ((size_t)p * 16u + l));

  if (tid == 0) out[(size_t)NPTS * 32u] = 0.0f;   // tv = 0.0
}

} // namespace

extern "C" void kernel_launch(void* const* d_in, const int* in_sizes, int n_in,
                              void* d_out, int out_size, void* d_ws, size_t ws_size,
                              hipStream_t stream) {
  (void)in_sizes; (void)n_in; (void)out_size; (void)d_ws; (void)ws_size;
  const float* pos     = (const float*)d_in[0];
  const float* latents = (const float*)d_in[1];
  const float* bound   = (const float*)d_in[2];
  float* out           = (float*)d_out;

  const unsigned nthreads = NPTS * LVLS;          // 8,388,608
  hashgrid_fwd<<<dim3(nthreads / 256u), dim3(256u), 0, stream>>>(
      pos, latents, bound, out);
}